// HiPPOLegS_74131135528969
// MI455X (gfx1250) — compile-verified
//
#include <hip/hip_runtime.h>
#include <hip/hip_bf16.h>

#define N_STATE   64
#define BATCH     64
#define SEQ_LEN   8192
#define CHUNK_LEN 64
#define N_CHUNKS  (SEQ_LEN / CHUNK_LEN)   // 128
#define N_BTILES  (BATCH / 16)            // 4

typedef __attribute__((ext_vector_type(2))) float v2f;
typedef __attribute__((ext_vector_type(8))) float v8f;

__device__ __forceinline__ v8f wmma_f32_16x16x4(v2f a, v2f b, v8f c) {
    // 8 args: (neg_a, A, neg_b, B, c_mod, C, reuse_a, reuse_b)
    return __builtin_amdgcn_wmma_f32_16x16x4_f32(
        false, a, false, b, (short)0, c, false, false);
}

// Chain of 16 WMMAs: D(16x16) = Cstate(16x64) @ W(64x16), Cstate read from LDS
// in [col][row] layout; A-fragment: lane holds M=lm, VGPR rr holds K=4kk+2h+rr.
__device__ __forceinline__ v8f state_matmul(const float* __restrict__ cb,
                                            const v2f (&bfrag)[17],
                                            int h, int lm) {
    v8f acc = {};
#pragma unroll
    for (int kk = 0; kk < 16; ++kk) {
        v2f a;
        a[0] = cb[(4 * kk + 2 * h + 0) * 16 + lm];
        a[1] = cb[(4 * kk + 2 * h + 1) * 16 + lm];
        acc = wmma_f32_16x16x4(a, bfrag[kk], acc);
    }
    return acc;
}

// Publish a D tile (acc[r] = C[M=r+8h][N=lm], n-tile wv) into [col][row] LDS:
// rows contiguous per lane -> two b128 ds stores.
__device__ __forceinline__ void publish_tile(float* __restrict__ cw_base,
                                             int wv, int lm, int h, v8f acc) {
    float* cw = cw_base + (16 * wv + lm) * 16 + 8 * h;
    *(float4*)(cw)     = make_float4(acc[0], acc[1], acc[2], acc[3]);
    *(float4*)(cw + 4) = make_float4(acc[4], acc[5], acc[6], acc[7]);
}

// Load resident B-operand fragments (W = rows of Ahat, cols 16wv..16wv+15):
// B(4x16) layout: lane holds N=lm, VGPR rr holds K=4kk+2h+rr.
__device__ __forceinline__ void load_bfrag17(const float* __restrict__ Ahat,
                                             v2f (&bfrag)[17],
                                             int wv, int lm, int h) {
#pragma unroll
    for (int kk = 0; kk < 17; ++kk) {
#pragma unroll
        for (int rr = 0; rr < 2; ++rr) {
            int k = 4 * kk + 2 * h + rr;
            bfrag[kk][rr] = Ahat[k * 64 + 16 * wv + lm];
        }
    }
}

// ---------------------------------------------------------------------------
// Kernel 1: bilinear discretization.  M = I - A/2 is lower triangular with
// positive diagonal -> forward substitution on [I + A/2 | B].
// Ahat (68x64): Ahat[k][n] = A_bar[n][k] (k<64); row 64 = B_bar; 65..67 = 0.
// ---------------------------------------------------------------------------
__global__ void hippo_discretize(const float* __restrict__ A,
                                 const float* __restrict__ B,
                                 float* __restrict__ Ahat) {
    __shared__ float Msh[64][64];
    __shared__ float Rsh[64][65];
    const int r = threadIdx.x;           // 0..63

    for (int c = 0; c < 64; ++c) {
        float a  = A[r * 64 + c];
        float id = (r == c) ? 1.0f : 0.0f;
        Msh[r][c] = id - 0.5f * a;
        Rsh[r][c] = id + 0.5f * a;
    }
    Rsh[r][64] = B[r];                   // dt = 1
    __syncthreads();

    for (int k = 0; k < 64; ++k) {
        if (r == k) {
            float inv = 1.0f / Msh[k][k];
            for (int c = 0; c <= 64; ++c) Rsh[k][c] *= inv;
        }
        __syncthreads();
        if (r > k) {
            float f = Msh[r][k];         // original value stays valid (tri)
            for (int c = 0; c <= 64; ++c) Rsh[r][c] -= f * Rsh[k][c];
        }
        __syncthreads();
    }

    for (int n = 0; n < 64; ++n) Ahat[r * 64 + n] = Rsh[n][r];
    Ahat[64 * 64 + r] = Rsh[r][64];
    Ahat[65 * 64 + r] = 0.0f;
    Ahat[66 * 64 + r] = 0.0f;
    Ahat[67 * 64 + r] = 0.0f;
}

// ---------------------------------------------------------------------------
// Kernel 2: A64hat[k][n] = (A_bar^64)[n][k] via 6 repeated squarings in LDS.
// ---------------------------------------------------------------------------
__global__ void hippo_matpow64(const float* __restrict__ Ahat,
                               float* __restrict__ A64hat) {
    __shared__ float P[64][64];
    __shared__ float Q[64][64];
    const int tid = threadIdx.x;         // 0..255
    const int row = tid >> 2;
    const int cb  = (tid & 3) * 16;

    for (int i = 0; i < 16; ++i)         // P[n][k] = A_bar[n][k]
        P[row][cb + i] = Ahat[(cb + i) * 64 + row];

    for (int it = 0; it < 6; ++it) {
        __syncthreads();
        for (int i = 0; i < 16; ++i) {
            float s = 0.0f;
            for (int k = 0; k < 64; ++k) s += P[row][k] * P[k][cb + i];
            Q[row][cb + i] = s;
        }
        __syncthreads();
        for (int i = 0; i < 16; ++i) P[row][cb + i] = Q[row][cb + i];
    }
    __syncthreads();
    for (int i = 0; i < 16; ++i)
        A64hat[(cb + i) * 64 + row] = P[row][cb + i];   // transpose out
}

// ---------------------------------------------------------------------------
// Pass A: per-chunk local scan from zero state; emit only the chunk-end state
// d_j (stored in D-fragment order, no layout conversion needed downstream).
// Grid: (N_CHUNKS, N_BTILES) x 128 threads (4 waves, one n-tile each).
// ---------------------------------------------------------------------------
__global__ void __launch_bounds__(128, 1)
hippo_chunk_local(const float* __restrict__ fin,
                  const float* __restrict__ Ahat,
                  float* __restrict__ dstate) {
    __shared__ __align__(16) float cbuf[2][N_STATE * 16];
    const int lane  = threadIdx.x & 31;
    const int wv    = threadIdx.x >> 5;
    const int h     = lane >> 4;
    const int lm    = lane & 15;
    const int chk   = blockIdx.x;
    const int btile = blockIdx.y;
    const int bbase = btile * 16;

    v2f bfrag[17];
    load_bfrag17(Ahat, bfrag, wv, lm, h);

    for (int i = threadIdx.x; i < N_STATE * 16; i += 128) cbuf[0][i] = 0.0f;
    __syncthreads();

    const float* frow = fin + (size_t)(bbase + lm) * SEQ_LEN + chk * CHUNK_LEN;
    float4 fq = make_float4(0.f, 0.f, 0.f, 0.f);
    v8f acc = {};

#pragma unroll 2
    for (int t = 0; t < CHUNK_LEN; ++t) {
        if ((t & 3) == 0) {
            fq = *(const float4*)(frow + t);
            if (t == 0 && chk + 1 < N_CHUNKS)
                __builtin_prefetch(frow + CHUNK_LEN, 0, 1);  // global_prefetch_b8
        }
        float fv = ((t & 3) == 0) ? fq.x
                 : ((t & 3) == 1) ? fq.y
                 : ((t & 3) == 2) ? fq.z : fq.w;

        acc = state_matmul(cbuf[t & 1], bfrag, h, lm);
        v2f af;                       // K-chunk 16: col 64 = f_t (B_bar row)
        af[0] = h ? 0.0f : fv;
        af[1] = 0.0f;
        acc = wmma_f32_16x16x4(af, bfrag[16], acc);

        publish_tile(cbuf[(t + 1) & 1], wv, lm, h, acc);
        __syncthreads();
    }

    float* dp = dstate + (((size_t)chk * N_BTILES + btile) * 4 + wv) * 256 + lane * 8;
    *(float4*)(dp)     = make_float4(acc[0], acc[1], acc[2], acc[3]);
    *(float4*)(dp + 4) = make_float4(acc[4], acc[5], acc[6], acc[7]);
}

// ---------------------------------------------------------------------------
// Combine: sequential scan over chunk boundaries (128 steps):
//   cstart[0] = 0;  cstart[j+1] = cstart[j] @ (A^64)^T + d[j]
// Records every cstart[j].  Grid: N_BTILES x 128 threads.
// ---------------------------------------------------------------------------
__global__ void __launch_bounds__(128, 1)
hippo_chunk_combine(const float* __restrict__ A64hat,
                    const float* __restrict__ dstate,
                    float* __restrict__ cstart) {
    __shared__ __align__(16) float cbuf[2][N_STATE * 16];
    const int lane  = threadIdx.x & 31;
    const int wv    = threadIdx.x >> 5;
    const int h     = lane >> 4;
    const int lm    = lane & 15;
    const int btile = blockIdx.x;

    v2f bfrag[16];
#pragma unroll
    for (int kk = 0; kk < 16; ++kk)
#pragma unroll
        for (int rr = 0; rr < 2; ++rr)
            bfrag[kk][rr] = A64hat[(4 * kk + 2 * h + rr) * 64 + 16 * wv + lm];

    for (int i = threadIdx.x; i < N_STATE * 16; i += 128) cbuf[0][i] = 0.0f;
    __syncthreads();

    v8f c = {};                          // cstart[0] = 0 (regs + cbuf[0])
    for (int j = 0; j < N_CHUNKS; ++j) {
        const size_t slot = (((size_t)j * N_BTILES + btile) * 4 + wv) * 256 + lane * 8;
        float4 d0 = *(const float4*)(dstate + slot);
        float4 d1 = *(const float4*)(dstate + slot + 4);

        // record cstart[j] (current state, pre-update)
        *(float4*)(cstart + slot)     = make_float4(c[0], c[1], c[2], c[3]);
        *(float4*)(cstart + slot + 4) = make_float4(c[4], c[5], c[6], c[7]);

        v8f nc = {};
        {
            const float* cb = cbuf[j & 1];
            v8f acc = state_matmul(cb, *(const v2f(*)[17])&bfrag, h, lm);
            // note: only first 16 entries of the reinterpreted array are read
            nc = acc;
        }
        nc[0] += d0.x; nc[1] += d0.y; nc[2] += d0.z; nc[3] += d0.w;
        nc[4] += d1.x; nc[5] += d1.y; nc[6] += d1.z; nc[7] += d1.w;
        c = nc;

        publish_tile(cbuf[(j + 1) & 1], wv, lm, h, c);
        __syncthreads();
    }
}

// ---------------------------------------------------------------------------
// Pass B: re-run each chunk's scan seeded with the true cstart[j], streaming
// the output (single write of the 128 MiB result -> HBM floor ~5.5 us).
// Grid: (N_CHUNKS, N_BTILES) x 128 threads.
// ---------------------------------------------------------------------------
__global__ void __launch_bounds__(128, 1)
hippo_chunk_final(const float* __restrict__ fin,
                  const float* __restrict__ Ahat,
                  const float* __restrict__ cstart,
                  float* __restrict__ out) {
    __shared__ __align__(16) float cbuf[2][N_STATE * 16];
    const int lane  = threadIdx.x & 31;
    const int wv    = threadIdx.x >> 5;
    const int h     = lane >> 4;
    const int lm    = lane & 15;
    const int chk   = blockIdx.x;
    const int btile = blockIdx.y;
    const int bbase = btile * 16;

    v2f bfrag[17];
    load_bfrag17(Ahat, bfrag, wv, lm, h);

    // seed cbuf[0] with cstart[chk] (same fragment order as it was stored)
    {
        const float* cp = cstart +
            (((size_t)chk * N_BTILES + btile) * 4 + wv) * 256 + lane * 8;
        float4 c0 = *(const float4*)(cp);
        float4 c1 = *(const float4*)(cp + 4);
        v8f s;
        s[0] = c0.x; s[1] = c0.y; s[2] = c0.z; s[3] = c0.w;
        s[4] = c1.x; s[5] = c1.y; s[6] = c1.z; s[7] = c1.w;
        publish_tile(cbuf[0], wv, lm, h, s);
    }
    __syncthreads();

    const float* frow = fin + (size_t)(bbase + lm) * SEQ_LEN + chk * CHUNK_LEN;
    float* obase = out + (size_t)bbase * SEQ_LEN * 64
                       + (size_t)chk * CHUNK_LEN * 64 + 16 * wv + lm;
    float4 fq = make_float4(0.f, 0.f, 0.f, 0.f);

#pragma unroll 2
    for (int t = 0; t < CHUNK_LEN; ++t) {
        if ((t & 3) == 0) fq = *(const float4*)(frow + t);
        float fv = ((t & 3) == 0) ? fq.x
                 : ((t & 3) == 1) ? fq.y
                 : ((t & 3) == 2) ? fq.z : fq.w;

        v8f acc = state_matmul(cbuf[t & 1], bfrag, h, lm);
        v2f af;
        af[0] = h ? 0.0f : fv;
        af[1] = 0.0f;
        acc = wmma_f32_16x16x4(af, bfrag[16], acc);

        publish_tile(cbuf[(t + 1) & 1], wv, lm, h, acc);

        float* op = obase + (size_t)t * 64;
#pragma unroll
        for (int rr = 0; rr < 8; ++rr)
            op[(size_t)(rr + 8 * h) * (SEQ_LEN * 64)] = acc[rr];

        __syncthreads();
    }
}

extern "C" void kernel_launch(void* const* d_in, const int* in_sizes, int n_in,
                              void* d_out, int out_size, void* d_ws, size_t ws_size,
                              hipStream_t stream) {
    const float* fin = (const float*)d_in[0];   // (64, 8192) f32
    const float* A   = (const float*)d_in[1];   // (64, 64) f32
    const float* B   = (const float*)d_in[2];   // (64,) f32
    float* out = (float*)d_out;                 // (64, 8192, 64) f32

    // Workspace layout (floats):
    float* ws     = (float*)d_ws;
    float* Ahat   = ws;                                  // 68*64   = 4352
    float* A64hat = ws + 4352;                           // 64*64   = 4096
    float* dstate = ws + 8448;                           // 128*4*1024 = 524288
    float* cst    = ws + 8448 + (size_t)N_CHUNKS * N_BTILES * 1024; // 524288

    hipLaunchKernelGGL(hippo_discretize, dim3(1), dim3(64), 0, stream, A, B, Ahat);
    hipLaunchKernelGGL(hippo_matpow64, dim3(1), dim3(256), 0, stream, Ahat, A64hat);
    hipLaunchKernelGGL(hippo_chunk_local, dim3(N_CHUNKS, N_BTILES), dim3(128),
                       0, stream, fin, Ahat, dstate);
    hipLaunchKernelGGL(hippo_chunk_combine, dim3(N_BTILES), dim3(128),
                       0, stream, A64hat, dstate, cst);
    hipLaunchKernelGGL(hippo_chunk_final, dim3(N_CHUNKS, N_BTILES), dim3(128),
                       0, stream, fin, Ahat, cst, out);
}